// WriteGate_35270271435147
// MI455X (gfx1250) — compile-verified
//
#include <hip/hip_runtime.h>

// WriteGate for MI455X (gfx1250): bandwidth-bound GEMV via V_WMMA_F32_16X16X4_F32,
// per-batch bitonic top-k in LDS, wide coalesced gather.

#define B_ 8
#define T_ 4096
#define H_ 2048
#define K_ 256
#define SLOT_ELEMS ((long long)B_ * K_ * H_)

typedef __attribute__((ext_vector_type(2))) float v2f;
typedef __attribute__((ext_vector_type(8))) float v8f;

// ---------------------------------------------------------------------------
// Phase 1: scores[b,t] = dot(hidden[b,t,:], W) + bias
// One wave32 computes 16 consecutive timesteps. A-matrix (16x4 f32): lanes 0-15
// hold rows M=0..15 with K slots {0,1}; lanes 16-31 hold K slots {2,3}.
// B-matrix: W chunk replicated across all 16 columns -> every D column holds
// the same 16 scores. We pick a slot->H mapping so each lane issues one
// b128 load per 8 H-elements (2 WMMAs), identical mapping for A and B.
// ---------------------------------------------------------------------------
__global__ __launch_bounds__(256) void wg_scores(const float* __restrict__ hidden,
                                                 const float* __restrict__ W,
                                                 const float* __restrict__ bias,
                                                 float* __restrict__ scores) {
  const int lane = threadIdx.x & 31;
  const int wave = (int)((blockIdx.x * blockDim.x + threadIdx.x) >> 5); // 0..2047
  const long long row0 = (long long)wave * 16;          // row index into flat (B*T)
  const int m = lane & 15;                              // matrix row within tile
  const int p = lane >> 4;                              // lane half (K-slot group)

  const float* rowPtr = hidden + (row0 + m) * (long long)H_;

  v8f c = {};
  for (int h = 0; h < H_; h += 8) {
    // lane half p covers H-elements [h+4p .. h+4p+3]
    float4 a4 = *(const float4*)(rowPtr + h + 4 * p);
    float4 b4 = *(const float4*)(W + h + 4 * p);
    v2f a0 = {a4.x, a4.y};
    v2f b0 = {b4.x, b4.y};
    v2f a1 = {a4.z, a4.w};
    v2f b1 = {b4.z, b4.w};
    // WMMA #0 accumulates H elems {h,h+1,h+4,h+5}; #1 accumulates {h+2,h+3,h+6,h+7}
    c = __builtin_amdgcn_wmma_f32_16x16x4_f32(false, a0, false, b0, (short)0, c, false, false);
    c = __builtin_amdgcn_wmma_f32_16x16x4_f32(false, a1, false, b1, (short)0, c, false, false);
  }

  // D layout (16x16 f32): VGPR r -> M=r (lanes 0-15), M=r+8 (lanes 16-31).
  // Column N = lane%16; every column identical, so lanes with m==0 write out.
  const float bb = bias[0];
  if (m == 0) {
    const long long base = row0 + (long long)p * 8;
#pragma unroll
    for (int r = 0; r < 8; ++r) scores[base + r] = c[r] + bb;
  }
}

// ---------------------------------------------------------------------------
// Phase 2: per-batch top-256 of 4096 scores via full bitonic sort in LDS
// (descending by score, tie -> smaller index first, matching top_k), then a
// 256-wide bitonic on the selected indices (ascending). Also writes the mask.
// ---------------------------------------------------------------------------
__device__ __forceinline__ bool kv_greater(float va, int ia, float vb, int ib) {
  return (va > vb) || (va == vb && ia < ib);
}

__global__ __launch_bounds__(1024) void wg_topk(const float* __restrict__ scores,
                                                int* __restrict__ top_idx,
                                                float* __restrict__ out) {
  __shared__ float s_val[T_];
  __shared__ int s_idx[T_];
  __shared__ int s_top[K_];

  const int b = blockIdx.x;
  const int tid = threadIdx.x;

  for (int i = tid; i < T_; i += blockDim.x) {
    s_val[i] = scores[(long long)b * T_ + i];
    s_idx[i] = i;
  }
  __syncthreads();

  // Full bitonic sort, final direction: descending by score.
  for (int size = 2; size <= T_; size <<= 1) {
    for (int stride = size >> 1; stride > 0; stride >>= 1) {
      for (int t = tid; t < T_ / 2; t += blockDim.x) {
        int i = ((t & ~(stride - 1)) << 1) | (t & (stride - 1));
        int j = i | stride;
        bool wantDesc = ((i & size) == 0);
        float vi = s_val[i], vj = s_val[j];
        int ii = s_idx[i], ij = s_idx[j];
        bool iFirst = kv_greater(vi, ii, vj, ij);
        if (iFirst != wantDesc) {
          s_val[i] = vj; s_val[j] = vi;
          s_idx[i] = ij; s_idx[j] = ii;
        }
      }
      __syncthreads();
    }
  }

  // Top-256 indices -> sort ascending.
  if (tid < K_) s_top[tid] = s_idx[tid];
  __syncthreads();

  for (int size = 2; size <= K_; size <<= 1) {
    for (int stride = size >> 1; stride > 0; stride >>= 1) {
      if (tid < K_ / 2) {
        int i = ((tid & ~(stride - 1)) << 1) | (tid & (stride - 1));
        int j = i | stride;
        bool wantAsc = ((i & size) == 0);
        int a = s_top[i], c = s_top[j];
        bool iFirst = (a < c);
        if (iFirst != wantAsc) { s_top[i] = c; s_top[j] = a; }
      }
      __syncthreads();
    }
  }

  if (tid < K_) {
    top_idx[b * K_ + tid] = s_top[tid];
    // mask: k == MEMORY_SLOTS, so all ones
    out[SLOT_ELEMS + (long long)b * K_ + tid] = 1.0f;
  }
}

// ---------------------------------------------------------------------------
// Phase 3: slots[b,i,:] = hidden[b, idx[b,i], :]  — coalesced float4 copy.
// One 256-thread block per output row (8 KB): 2 float4 per thread.
// ---------------------------------------------------------------------------
__global__ __launch_bounds__(256) void wg_gather(const float* __restrict__ hidden,
                                                 const int* __restrict__ top_idx,
                                                 float* __restrict__ out) {
  const int row = blockIdx.x;        // 0 .. B_*K_-1
  const int b = row >> 8;
  const int t = top_idx[row];
  const float4* src = (const float4*)(hidden + ((long long)b * T_ + t) * (long long)H_);
  float4* dst = (float4*)(out + (long long)row * H_);
#pragma unroll
  for (int c = threadIdx.x; c < H_ / 4; c += 256) {
    dst[c] = src[c];
  }
}

extern "C" void kernel_launch(void* const* d_in, const int* in_sizes, int n_in,
                              void* d_out, int out_size, void* d_ws, size_t ws_size,
                              hipStream_t stream) {
  const float* hidden = (const float*)d_in[0];
  const float* W      = (const float*)d_in[1];
  const float* bias   = (const float*)d_in[2];
  float* out          = (float*)d_out;

  float* scores  = (float*)d_ws;                                   // B*T floats = 128 KB
  int*   top_idx = (int*)((char*)d_ws + (size_t)B_ * T_ * sizeof(float)); // 8 KB

  // Phase 1: (B*T/16) waves, 8 waves per 256-thread block -> 256 blocks.
  wg_scores<<<(B_ * T_ / 16) / 8, 256, 0, stream>>>(hidden, W, bias, scores);
  // Phase 2: one block per batch.
  wg_topk<<<B_, 1024, 0, stream>>>(scores, top_idx, out);
  // Phase 3: one block per gathered row.
  wg_gather<<<B_ * K_, 256, 0, stream>>>(hidden, top_idx, out);
}